// MindPalaceRouter_6631429505423
// MI455X (gfx1250) — compile-verified
//
#include <hip/hip_runtime.h>
#include <math.h>

typedef float v2f __attribute__((ext_vector_type(2)));
typedef float v8f __attribute__((ext_vector_type(8)));

#define D_MODEL 2048
#define N_ROOMS 128
#define BSZ     128
#define SEQ     512
#define KC      64      // K-chunk staged in LDS
#define LDA_PAD 68      // 64 + 4: multiple of 4 (16B stores), stride%64=4 -> conflict-free

// ---------------- Kernel 1: partial sums over S (bandwidth kernel) ----------------
// grid: (D_MODEL/4/256, B, 4), block 256. partial[c][b][d] = sum_{s in chunk c} x[b,s,d]
__launch_bounds__(256)
__global__ void k_partial_mean(const float* __restrict__ x, float* __restrict__ partial) {
    int d4 = blockIdx.x * blockDim.x + threadIdx.x;       // 0..511 (float4 index in D)
    int b  = blockIdx.y;
    int c  = blockIdx.z;                                   // S chunk, 128 each
    const float4* xp = (const float4*)x
        + (size_t)b * (SEQ * (D_MODEL / 4))
        + (size_t)(c * 128) * (D_MODEL / 4) + d4;
    float4 acc = make_float4(0.f, 0.f, 0.f, 0.f);
    #pragma unroll 4
    for (int s = 0; s < 128; ++s) {
        float4 v = xp[(size_t)s * (D_MODEL / 4)];
        acc.x += v.x; acc.y += v.y; acc.z += v.z; acc.w += v.w;
    }
    ((float4*)partial)[((size_t)c * BSZ + b) * (D_MODEL / 4) + d4] = acc;
}

// ---------------- Kernel 2: finalize mean ----------------
__launch_bounds__(256)
__global__ void k_finalize_mean(const float* __restrict__ partial, float* __restrict__ xmean) {
    int i = blockIdx.x * blockDim.x + threadIdx.x;         // over B*D_MODEL/4 float4s
    const int stride = BSZ * D_MODEL / 4;
    const float4* p = (const float4*)partial;
    float4 a = p[i], b = p[i + stride], c = p[i + 2 * stride], d = p[i + 3 * stride];
    const float inv = 1.0f / (float)SEQ;
    float4 r;
    r.x = (a.x + b.x + c.x + d.x) * inv;
    r.y = (a.y + b.y + c.y + d.y) * inv;
    r.z = (a.z + b.z + c.z + d.z) * inv;
    r.w = (a.w + b.w + c.w + d.w) * inv;
    ((float4*)xmean)[i] = r;
}

// ---------------- Kernel 3: fp32 WMMA GEMM: C[128,N] = alpha*(A[128,K] @ W[N,K]^T + bias) ----
// grid.x = N/128, block = 256 (8 waves). Wave w -> cols [n0+16w, n0+16w+16), all 128 rows.
__launch_bounds__(256)
__global__ void k_gemm_wmma(const float* __restrict__ A,     // [128, K] row-major
                            const float* __restrict__ W,     // [N, K]   row-major
                            const float* __restrict__ bias,  // [N] or nullptr
                            float* __restrict__ C,           // [128, N]
                            int N, int K, float alpha) {
    __shared__ __align__(16) float As[128 * LDA_PAD];

    const int tid  = threadIdx.x;
    const int wave = tid >> 5;
    const int lane = tid & 31;
    const int l16  = lane & 15;
    const int h    = lane >> 4;            // half-wave selector (K offset 0 or 2)
    const int n0   = blockIdx.x * 128;
    const int ncol = n0 + wave * 16 + l16;

    v8f acc[8];
    #pragma unroll
    for (int m = 0; m < 8; ++m) { v8f z = {}; acc[m] = z; }

    const float* wrow = W + (size_t)ncol * K;

    for (int kc = 0; kc < K; kc += KC) {
        __syncthreads();   // protect previous chunk's readers
        // cooperatively stage A[0:128, kc:kc+64] -> LDS (float4 per thread, 8 iters)
        for (int t = tid; t < 128 * (KC / 4); t += 256) {
            int row = t >> 4;
            int c4  = (t & 15) << 2;
            float4 v = *(const float4*)(A + (size_t)row * K + kc + c4);
            *(float4*)(&As[row * LDA_PAD + c4]) = v;
        }
        __syncthreads();

        if (kc + KC < K) __builtin_prefetch(wrow + kc + KC + 2 * h, 0, 1);

        for (int kk = 0; kk < KC; kk += 4) {
            // B fragment: b[v] = W[ncol][kc+kk+2h+v]  (K = v + 2h lane layout)
            v2f bfrag = *(const v2f*)(wrow + kc + kk + 2 * h);

            // Batch all 8 A-fragment LDS loads into distinct regs so the
            // DS pipeline keeps multiple loads in flight (single dscnt wait
            // instead of 8 serialized load->wait->wmma round trips).
            v2f af[8];
            #pragma unroll
            for (int m = 0; m < 8; ++m)
                af[m] = *(const v2f*)(&As[(m * 16 + l16) * LDA_PAD + kk + 2 * h]);

            #pragma unroll
            for (int m = 0; m < 8; ++m)
                acc[m] = __builtin_amdgcn_wmma_f32_16x16x4_f32(
                    false, af[m], false, bfrag, (short)0, acc[m], false, false);
        }
    }

    const float bv = (bias != nullptr) ? bias[ncol] : 0.0f;
    #pragma unroll
    for (int m = 0; m < 8; ++m) {
        #pragma unroll
        for (int r = 0; r < 8; ++r) {
            int row = m * 16 + r + 8 * h;   // C/D layout: VGPR r -> M=r (+8 for upper lanes)
            C[(size_t)row * N + ncol] = alpha * (acc[m][r] + bv);
        }
    }
}

// ---------------- Kernel 4: softmax(adj + warp) @ raw_scores -> sigmoid ----------------
// one wave32 per (b,i) row of 128; block = 8 waves; grid = B*N/8
__launch_bounds__(256)
__global__ void k_route(const float* __restrict__ warp_scaled,  // [B*N, N] = 0.1*(ctx@Ww.T+bw)
                        const float* __restrict__ adjacency,    // [N, N]
                        const float* __restrict__ raw_scores,   // [B, N]
                        float* __restrict__ out) {              // [B, N]
    const int wave = threadIdx.x >> 5;
    const int lane = threadIdx.x & 31;
    const int row  = blockIdx.x * 8 + wave;   // 0 .. B*N-1
    const int b    = row >> 7;
    const int i    = row & 127;

    const float* wr = warp_scaled + (size_t)row * N_ROOMS;
    const float* ar = adjacency + (size_t)i * N_ROOMS;
    const float* rs = raw_scores + (size_t)b * N_ROOMS;

    float v[4];
    float mx = -3.402823466e38f;
    #pragma unroll
    for (int t = 0; t < 4; ++t) {
        int j = lane + 32 * t;
        v[t] = ar[j] + wr[j];
        mx = fmaxf(mx, v[t]);
    }
    #pragma unroll
    for (int off = 16; off > 0; off >>= 1) mx = fmaxf(mx, __shfl_xor(mx, off, 32));

    float sum = 0.f, dot = 0.f;
    #pragma unroll
    for (int t = 0; t < 4; ++t) {
        int j = lane + 32 * t;
        float e = __expf(v[t] - mx);
        sum += e;
        dot += e * rs[j];
    }
    #pragma unroll
    for (int off = 16; off > 0; off >>= 1) {
        sum += __shfl_xor(sum, off, 32);
        dot += __shfl_xor(dot, off, 32);
    }
    if (lane == 0) {
        float wsv = dot / sum;
        out[row] = 1.0f / (1.0f + __expf(-wsv));
    }
}

extern "C" void kernel_launch(void* const* d_in, const int* in_sizes, int n_in,
                              void* d_out, int out_size, void* d_ws, size_t ws_size,
                              hipStream_t stream) {
    const float* x    = (const float*)d_in[0];   // [B,S,D]
    const float* room = (const float*)d_in[1];   // [N,D]
    const float* Wc   = (const float*)d_in[2];   // [D,D]
    const float* bc   = (const float*)d_in[3];   // [D]
    const float* Ww   = (const float*)d_in[4];   // [N*N, D]
    const float* bw   = (const float*)d_in[5];   // [N*N]
    const float* adj  = (const float*)d_in[6];   // [N,N]
    float* out = (float*)d_out;

    float* ws      = (float*)d_ws;
    float* partial = ws;                                     // 4*B*D
    float* xmean   = partial + 4 * (size_t)BSZ * D_MODEL;    // B*D
    float* context = xmean   + (size_t)BSZ * D_MODEL;        // B*D
    float* warp    = context + (size_t)BSZ * D_MODEL;        // B*N*N
    float* raw     = warp    + (size_t)BSZ * N_ROOMS * N_ROOMS; // B*N

    dim3 g1(D_MODEL / 4 / 256, BSZ, 4);
    k_partial_mean<<<g1, 256, 0, stream>>>(x, partial);
    k_finalize_mean<<<(BSZ * D_MODEL / 4) / 256, 256, 0, stream>>>(partial, xmean);

    k_gemm_wmma<<<D_MODEL / 128, 256, 0, stream>>>(xmean, Wc, bc, context,
                                                   D_MODEL, D_MODEL, 1.0f);
    k_gemm_wmma<<<(N_ROOMS * N_ROOMS) / 128, 256, 0, stream>>>(context, Ww, bw, warp,
                                                   N_ROOMS * N_ROOMS, D_MODEL, 0.1f);
    k_gemm_wmma<<<N_ROOMS / 128, 256, 0, stream>>>(context, room, nullptr, raw,
                                                   N_ROOMS, D_MODEL, 1.0f);

    k_route<<<(BSZ * N_ROOMS) / 8, 256, 0, stream>>>(warp, adj, raw, out);
}